// FNetBlock_18923625906214
// MI455X (gfx1250) — compile-verified
//
#include <hip/hip_runtime.h>
#include <hip/hip_bf16.h>
#include <math.h>

// FNet mixing as DFT-by-GEMM on CDNA5 (gfx1250):
//   A = x @ Ch ; B = x @ Sh            (stage 1, hidden-dim DFT, H=768)
//   out = Cseq @ A - Sseq @ B          (stage 2, seq-dim DFT, S=4096)
// f32 WMMA 16x16x4; twiddle phases maintained as exact integer arithmetic
// progressions mod N (one add + cond-subtract per step, no div/mod in the
// hot loop); tiles staged into LDS by the Tensor Data Mover with double
// buffering (TENSORcnt waits); one twiddle fragment feeds four WMMAs.

namespace {
constexpr int kB = 8;      // batch
constexpr int kS = 4096;   // sequence length
constexpr int kH = 768;    // hidden dim

typedef __attribute__((ext_vector_type(2))) float v2f;
typedef __attribute__((ext_vector_type(8))) float v8f;
typedef __attribute__((ext_vector_type(4))) unsigned u32x4;
typedef __attribute__((ext_vector_type(8))) int i32x8;
typedef __attribute__((ext_vector_type(4))) int i32x4;

// ---------------------------------------------------------------------------
// TDM: 2-D tile load (global -> LDS), 4-byte elements.
//   word0 carries data_size / LDS-padding config (see D# group1 layout).
//   6-arg builtin: (u32x4 g0, i32x8 g1, i32x4 g2, i32x4 g3, i32x8 g4, i32 cpol)
// ---------------------------------------------------------------------------
__device__ __forceinline__ void tdm_load_2d(const void* gaddr, unsigned lds_off,
                                            unsigned tile_w, unsigned tile_h,
                                            unsigned tensor_w, unsigned tensor_h,
                                            unsigned stride_elems,
                                            unsigned word0) {
  unsigned long long ga = (unsigned long long)(uintptr_t)gaddr;
  u32x4 g0;
  g0[0] = 1u;                                              // count=1 (valid D#)
  g0[1] = lds_off;                                         // LDS byte address
  g0[2] = (unsigned)(ga & 0xffffffffu);                    // global_addr lo
  g0[3] = (unsigned)((ga >> 32) & 0x1ffffffu) | (2u << 30);// global hi | type=2
  i32x8 g1;
  g1[0] = (int)word0;                                      // mask=0|data_size|pad
  g1[1] = (int)((tensor_w & 0xffffu) << 16);               // tensor_dim0 lo16
  g1[2] = (int)(((tensor_w >> 16) & 0xffffu) |
                ((tensor_h & 0xffffu) << 16));             // dim0 hi | dim1 lo
  g1[3] = (int)(((tensor_h >> 16) & 0xffffu) |
                ((tile_w & 0xffffu) << 16));               // dim1 hi | tile_dim0
  g1[4] = (int)(tile_h & 0xffffu);                         // tile_dim1 (dim2=0)
  g1[5] = (int)stride_elems;                               // dim0 stride lo32
  g1[6] = 0;                                               // stride hi / dim1 stride
  g1[7] = 0;
  i32x4 z4 = {};
  i32x8 z8 = {};
  __builtin_amdgcn_tensor_load_to_lds(g0, g1, z4, z4, z8, 0);
}

// data_size=4B (code 2 at bits[17:16])
constexpr unsigned kTdmPlain = 2u << 16;
// + pad_enable(bit20), pad_interval code 4 => every 32 DWORDs (bits[24:22]),
//   pad_amount code 0 => 1 DWORD  -> LDS row stride 33 floats
constexpr unsigned kTdmPad33 = (2u << 16) | (1u << 20) | (4u << 22);

// ---------------------------------------------------------------------------
// Stage 1: rows r = (b,t) flattened (32768), cols h (768).
//   A[r,h] = sum_k x[r,k] cos(2*pi*k*h/H);  B[r,h] = sum_k x[r,k] sin(...)
// Block 256 thr = 8 waves as 2(M)x4(N); wave tile = 32(M)x16(N): one twiddle
// B-fragment (2 sincos) feeds 4 WMMAs. Block tile 64M x 64N.
// x tile 64x32 staged by TDM with pad -> LDS stride 33 (conflict-free, since
// A-fragment reads vary by *row* across lanes).
// Twiddle phase p = (k*n) mod H kept incrementally: k steps by 4 uniformly,
// so p += (4n mod H) with one conditional subtract (no div/mod in the loop).
// ---------------------------------------------------------------------------
__global__ __launch_bounds__(256) void fnet_stage1(const float* __restrict__ x,
                                                   float* __restrict__ Aout,
                                                   float* __restrict__ Bout) {
  __shared__ float xs[2][64][33];

  const int tid   = threadIdx.x;
  const int wave  = tid >> 5;
  const int lane  = tid & 31;
  const int waveM = wave & 1;    // 0..1 -> 32-row slices
  const int waveN = wave >> 1;   // 0..3 -> 16-col slices
  const int l15   = lane & 15;
  const int lhi   = lane >> 4;

  const int rowBase = blockIdx.x * 64;
  const int n       = blockIdx.y * 64 + waveN * 16 + l15;  // twiddle col (h)
  const int m0      = waveM * 32 + l15;                    // LDS row, tile 0
  const int m1      = m0 + 16;                             // LDS row, tile 1

  const float w0 = 6.28318530717958647692f / (float)kH;

  v8f accC0 = {}, accS0 = {}, accC1 = {}, accS1 = {};

  // Incremental phase state (exact): p = (k*n) mod H at k = lane-half base.
  int phase  = (2 * lhi * n) % kH;     // one-time mod at setup
  const int dstep = (4 * n) % kH;      // phase advance per k-step (k += 4)

  // Preload first K-chunk.
  if (wave == 0) {
    tdm_load_2d(x + (size_t)rowBase * kH, (unsigned)(uintptr_t)&xs[0][0][0],
                32, 64, kH, kB * kS, kH, kTdmPad33);
  }

  int buf = 0;
  for (int k0 = 0; k0 < kH; k0 += 32) {
    if (wave == 0) {
      if (k0 + 32 < kH) {
        tdm_load_2d(x + (size_t)rowBase * kH + (k0 + 32),
                    (unsigned)(uintptr_t)&xs[buf ^ 1][0][0],
                    32, 64, kH, kB * kS, kH, kTdmPad33);
        __builtin_amdgcn_s_wait_tensorcnt((short)1);  // current tile done
      } else {
        __builtin_amdgcn_s_wait_tensorcnt((short)0);
      }
    }
    __syncthreads();

    #pragma unroll 4
    for (int kk = 0; kk < 32; kk += 4) {
      // Twiddle B-fragments from incremental phase: rows k, k+1 for this
      // lane half (k = k0+kk+2*lhi), col n.
      const int p0 = phase;
      int p1 = phase + n;           // (k+1)*n mod H, n < H -> one cond-sub
      if (p1 >= kH) p1 -= kH;
      phase += dstep;               // advance to next k-step
      if (phase >= kH) phase -= kH;

      float s0, c0, s1, c1;
      __sincosf(w0 * (float)p0, &s0, &c0);
      __sincosf(w0 * (float)p1, &s1, &c1);
      v2f bc, bsn;
      bc.x  = c0; bc.y  = c1;
      bsn.x = s0; bsn.y = s1;

      // x A-fragments: lane = row, VGPR0/1 = K pair.
      v2f a0, a1;
      a0.x = xs[buf][m0][kk + lhi * 2 + 0];
      a0.y = xs[buf][m0][kk + lhi * 2 + 1];
      a1.x = xs[buf][m1][kk + lhi * 2 + 0];
      a1.y = xs[buf][m1][kk + lhi * 2 + 1];

      accC0 = __builtin_amdgcn_wmma_f32_16x16x4_f32(false, a0, false, bc,
                                                    (short)0, accC0, false, false);
      accS0 = __builtin_amdgcn_wmma_f32_16x16x4_f32(false, a0, false, bsn,
                                                    (short)0, accS0, false, false);
      accC1 = __builtin_amdgcn_wmma_f32_16x16x4_f32(false, a1, false, bc,
                                                    (short)0, accC1, false, false);
      accS1 = __builtin_amdgcn_wmma_f32_16x16x4_f32(false, a1, false, bsn,
                                                    (short)0, accS1, false, false);
    }
    __syncthreads();
    buf ^= 1;
  }

  // C/D layout: c[r] -> row (r + 8*(lane>=16)), col (lane&15).
  const int row0 = rowBase + waveM * 32 + lhi * 8;
  #pragma unroll
  for (int r = 0; r < 8; ++r) {
    size_t i0 = (size_t)(row0 + r) * kH + n;
    size_t i1 = (size_t)(row0 + 16 + r) * kH + n;
    Aout[i0] = accC0[r];
    Bout[i0] = accS0[r];
    Aout[i1] = accC1[r];
    Bout[i1] = accS1[r];
  }
}

// ---------------------------------------------------------------------------
// Stage 2 (per batch): out[s,h] = sum_t cos(w s t) A[t,h] - sin(w s t) B[t,h]
// Block 256 thr = 8 waves as 4(M)x2(N); wave tile = 16(M)x32(N): one seq
// twiddle A-fragment (2 sincos, -sin folded) feeds 4 WMMAs. Block tile
// 64M x 64N. A/B intermediate tiles (32t x 64h) staged by TDM, packed
// stride 64 (fragment reads vary by *column* across lanes -> conflict-free).
// Phase p = (s*t) & (S-1) kept incrementally: p += (4s) & (S-1) per k-step.
// ---------------------------------------------------------------------------
__global__ __launch_bounds__(256) void fnet_stage2(const float* __restrict__ Ain,
                                                   const float* __restrict__ Bin,
                                                   float* __restrict__ out) {
  __shared__ float as[2][32][64];
  __shared__ float bs[2][32][64];

  const int tid   = threadIdx.x;
  const int wave  = tid >> 5;
  const int lane  = tid & 31;
  const int waveM = wave & 3;    // 0..3 -> 16-row slices (s)
  const int waveN = wave >> 2;   // 0..1 -> 32-col slices (h)
  const int l15   = lane & 15;
  const int lhi   = lane >> 4;

  const int batch   = blockIdx.z;
  const int rowBase = blockIdx.x * 64;            // s tile
  const int colBase = blockIdx.y * 64;            // h tile (block)
  const int cc0     = waveN * 32 + l15;           // LDS col, tile 0
  const int cc1     = cc0 + 16;                   // LDS col, tile 1
  const int srow    = rowBase + waveM * 16 + l15; // s index for A-frag lane

  const float* __restrict__ Ab = Ain + (size_t)batch * kS * kH;
  const float* __restrict__ Bb = Bin + (size_t)batch * kS * kH;

  const float w0 = 6.28318530717958647692f / (float)kS;
  constexpr unsigned kMask = (unsigned)(kS - 1);

  // Incremental phase state (exact, power-of-two modulus).
  unsigned phase = ((unsigned)srow * (unsigned)(2 * lhi)) & kMask;
  const unsigned dstep = ((unsigned)srow << 2) & kMask;   // t += 4
  const unsigned dp1   = (unsigned)srow & kMask;          // t += 1

  v8f acc0 = {}, acc1 = {};

  if (wave == 0) {
    tdm_load_2d(Ab + colBase, (unsigned)(uintptr_t)&as[0][0][0],
                64, 32, kH, kS, kH, kTdmPlain);
    tdm_load_2d(Bb + colBase, (unsigned)(uintptr_t)&bs[0][0][0],
                64, 32, kH, kS, kH, kTdmPlain);
  }

  int buf = 0;
  for (int k0 = 0; k0 < kS; k0 += 32) {
    if (wave == 0) {
      if (k0 + 32 < kS) {
        tdm_load_2d(Ab + (size_t)(k0 + 32) * kH + colBase,
                    (unsigned)(uintptr_t)&as[buf ^ 1][0][0],
                    64, 32, kH, kS, kH, kTdmPlain);
        tdm_load_2d(Bb + (size_t)(k0 + 32) * kH + colBase,
                    (unsigned)(uintptr_t)&bs[buf ^ 1][0][0],
                    64, 32, kH, kS, kH, kTdmPlain);
        __builtin_amdgcn_s_wait_tensorcnt((short)2);  // current 2 loads done
      } else {
        __builtin_amdgcn_s_wait_tensorcnt((short)0);
      }
    }
    __syncthreads();

    #pragma unroll 4
    for (int kk = 0; kk < 32; kk += 4) {
      const unsigned p0 = phase;
      const unsigned p1 = (phase + dp1) & kMask;
      phase = (phase + dstep) & kMask;

      float s0, c0, s1, c1;
      __sincosf(w0 * (float)p0, &s0, &c0);
      __sincosf(w0 * (float)p1, &s1, &c1);
      v2f ca, nsa;
      ca.x  = c0;  ca.y  = c1;
      nsa.x = -s0; nsa.y = -s1;   // fold the minus sign of the sin term

      v2f af0, bf0, af1, bf1;
      af0.x = as[buf][kk + lhi * 2 + 0][cc0];
      af0.y = as[buf][kk + lhi * 2 + 1][cc0];
      bf0.x = bs[buf][kk + lhi * 2 + 0][cc0];
      bf0.y = bs[buf][kk + lhi * 2 + 1][cc0];
      af1.x = as[buf][kk + lhi * 2 + 0][cc1];
      af1.y = as[buf][kk + lhi * 2 + 1][cc1];
      bf1.x = bs[buf][kk + lhi * 2 + 0][cc1];
      bf1.y = bs[buf][kk + lhi * 2 + 1][cc1];

      acc0 = __builtin_amdgcn_wmma_f32_16x16x4_f32(false, ca, false, af0,
                                                   (short)0, acc0, false, false);
      acc0 = __builtin_amdgcn_wmma_f32_16x16x4_f32(false, nsa, false, bf0,
                                                   (short)0, acc0, false, false);
      acc1 = __builtin_amdgcn_wmma_f32_16x16x4_f32(false, ca, false, af1,
                                                   (short)0, acc1, false, false);
      acc1 = __builtin_amdgcn_wmma_f32_16x16x4_f32(false, nsa, false, bf1,
                                                   (short)0, acc1, false, false);
    }
    __syncthreads();
    buf ^= 1;
  }

  float* __restrict__ ob = out + (size_t)batch * kS * kH;
  const int row0 = rowBase + waveM * 16 + lhi * 8;
  const int h0   = colBase + cc0;
  const int h1   = colBase + cc1;
  #pragma unroll
  for (int r = 0; r < 8; ++r) {
    ob[(size_t)(row0 + r) * kH + h0] = acc0[r];
    ob[(size_t)(row0 + r) * kH + h1] = acc1[r];
  }
}

}  // namespace

extern "C" void kernel_launch(void* const* d_in, const int* in_sizes, int n_in,
                              void* d_out, int out_size, void* d_ws, size_t ws_size,
                              hipStream_t stream) {
  (void)in_sizes; (void)n_in; (void)out_size; (void)ws_size;

  const float* x = (const float*)d_in[0];
  float* out = (float*)d_out;

  // Workspace: A intermediate (96 MiB) then B intermediate (96 MiB), f32.
  float* Aws = (float*)d_ws;
  float* Bws = Aws + (size_t)kB * kS * kH;

  dim3 g1((kB * kS) / 64, kH / 64);        // 512 x 12 blocks
  fnet_stage1<<<g1, 256, 0, stream>>>(x, Aws, Bws);

  dim3 g2(kS / 64, kH / 64, kB);           // 64 x 12 x 8 blocks
  fnet_stage2<<<g2, 256, 0, stream>>>(Aws, Bws, out);
}